// _proposal_71425306132562
// MI455X (gfx1250) — compile-verified
//
#include <hip/hip_runtime.h>
#include <hip/hip_bf16.h>
#include <float.h>

typedef __attribute__((ext_vector_type(2))) float v2f;
typedef __attribute__((ext_vector_type(8))) float v8f;

#define NMS_T       0.7f
#define POST_NMS    300
#define MAXN        2048
#define NMS_THREADS 1024

#if __has_builtin(__builtin_amdgcn_global_load_async_to_lds_b32) && \
    __has_builtin(__builtin_amdgcn_s_wait_asynccnt)
#define USE_ASYNC_LDS 1
#else
#define USE_ASYNC_LDS 0
#endif

typedef __attribute__((address_space(1))) int gas_int;   // global (as1)
typedef __attribute__((address_space(3))) int las_int;   // LDS (as3)

// ---------------------------------------------------------------------------
// Kernel 1: center->corner + clip, elementwise over all B*N boxes.
// ---------------------------------------------------------------------------
__global__ void clip_kernel(const float* __restrict__ rois,
                            const int* __restrict__ ph, const int* __restrict__ pw,
                            float* __restrict__ out, int total) {
  int i = blockIdx.x * blockDim.x + threadIdx.x;
  if (i >= total) return;
  float4 r = ((const float4*)rois)[i];
  float xmax = (float)(*pw) - 1.0f;
  float ymax = (float)(*ph) - 1.0f;
  float hw = 0.5f * r.z, hh = 0.5f * r.w;
  float4 o;
  o.x = fminf(fmaxf(r.x - hw, 0.0f), xmax);
  o.y = fminf(fmaxf(r.y - hh, 0.0f), ymax);
  o.z = fminf(fmaxf(r.x + hw, 0.0f), xmax);
  o.w = fminf(fmaxf(r.y + hh, 0.0f), ymax);
  ((float4*)out)[i] = o;
}

// ---------------------------------------------------------------------------
// Kernel 2: per-batch chunked greedy NMS. One 1024-thread block per batch.
// ---------------------------------------------------------------------------
__global__ __launch_bounds__(NMS_THREADS) void nms_kernel(
    const float* __restrict__ scores, const float* __restrict__ rois,
    const int* __restrict__ ph, const int* __restrict__ pw,
    int* __restrict__ keepOut, int B, int N)
{
  __shared__ float4 boxesS[MAXN];     // sorted clipped boxes
  __shared__ float  areaS[MAXN];      // areas of sorted boxes
  __shared__ float  skey[MAXN];       // sort keys (scores)
  __shared__ int    sidxs[MAXN];      // original indices, sorted order
  __shared__ unsigned rm[MAXN / 32];  // suppression bitmask
  __shared__ float4 chunkBox[16];
  __shared__ float  chunkArea[16];
  __shared__ int    chunkPos[16];
  __shared__ int      sh_chunkN, sh_base, sh_keepCount;
  __shared__ unsigned sh_alive;

  const int b   = blockIdx.x;
  const int tid = threadIdx.x;
  const int bs  = blockDim.x;
  if (N > MAXN) return;

  const float xmax = (float)(*pw) - 1.0f;
  const float ymax = (float)(*ph) - 1.0f;
  const float* sc    = scores + (size_t)b * N;
  const float* roisB = rois + (size_t)b * N * 4;

  // ---- init sort arrays (scores streamed via async global->LDS if available),
  //      and prefetch this batch's rois so the post-sort gather hits cache ----
  for (int i = tid; i < MAXN; i += bs) {
    if (i < N) {
#if USE_ASYNC_LDS
      __builtin_amdgcn_global_load_async_to_lds_b32((gas_int*)(sc + i),
                                                    (las_int*)&skey[i], 0, 0);
#else
      skey[i] = sc[i];
#endif
      sidxs[i] = i;
    } else {
      skey[i]  = -FLT_MAX;
      sidxs[i] = 0x7FFFFFFF;
    }
  }
  {
    int lines = (N * 16) >> 7;          // 128B cache lines covering rois[b]
    for (int l = tid; l < lines; l += bs)
      __builtin_prefetch(roisB + l * 32, 0, 0);
  }
#if USE_ASYNC_LDS
  __builtin_amdgcn_s_wait_asynccnt(0);
#endif
  __syncthreads();

  // ---- bitonic sort: descending score, ascending index on ties ----
  for (int k = 2; k <= MAXN; k <<= 1) {
    for (int j = k >> 1; j > 0; j >>= 1) {
      for (int idx = tid; idx < MAXN; idx += bs) {
        int ixj = idx ^ j;
        if (ixj > idx) {
          bool up = ((idx & k) == 0);
          float ka = skey[idx], kb = skey[ixj];
          int   ia = sidxs[idx], ib = sidxs[ixj];
          bool bBeforeA = (kb > ka) || ((kb == ka) && (ib < ia));
          if (bBeforeA == up) {
            skey[idx] = kb;  skey[ixj] = ka;
            sidxs[idx] = ib; sidxs[ixj] = ia;
          }
        }
      }
      __syncthreads();
    }
  }

  // ---- gather sorted boxes (recompute clip from rois; no global round-trip) ----
  for (int p = tid; p < MAXN; p += bs) {
    float4 o = make_float4(0.f, 0.f, 0.f, 0.f);
    if (p < N) {
      int j = sidxs[p];
      float4 r = ((const float4*)roisB)[j];
      float hw = 0.5f * r.z, hh = 0.5f * r.w;
      o.x = fminf(fmaxf(r.x - hw, 0.0f), xmax);
      o.y = fminf(fmaxf(r.y - hh, 0.0f), ymax);
      o.z = fminf(fmaxf(r.x + hw, 0.0f), xmax);
      o.w = fminf(fmaxf(r.y + hh, 0.0f), ymax);
    }
    boxesS[p] = o;
    areaS[p]  = (o.z - o.x) * (o.w - o.y);
  }
  for (int i = tid; i < MAXN / 32; i += bs) rm[i] = 0u;
  if (tid == 0) { sh_base = 0; sh_keepCount = 0; }
  __syncthreads();

  const int lane = tid & 31;
  const int wv   = tid >> 5;
  const int ncol = lane & 15;
  const int half = lane >> 4;

  // ---- chunked greedy NMS rounds ----
  for (;;) {
    // scan for next <=16 unsuppressed candidates (serial on t0, LDS only)
    if (tid == 0) {
      int cn = 0, p = sh_base;
      while (p < N && cn < 16) {
        if (!((rm[p >> 5] >> (p & 31)) & 1u)) {
          chunkPos[cn]  = p;
          chunkBox[cn]  = boxesS[p];
          chunkArea[cn] = areaS[p];
          cn++;
        }
        ++p;
      }
      for (int c = cn; c < 16; ++c) {
        chunkBox[c]  = make_float4(0.f, 0.f, 0.f, 0.f);
        chunkArea[c] = 0.f;
        chunkPos[c]  = 0;
      }
      sh_chunkN = cn;
      sh_base   = p;
    }
    __syncthreads();
    int cn = __builtin_amdgcn_readfirstlane(sh_chunkN);
    if (cn == 0) break;

    // intra-chunk greedy resolve (<=256 IoUs, serial, exact ordering)
    if (tid == 0) {
      unsigned alive = (cn >= 16) ? 0xFFFFu : ((1u << cn) - 1u);
      for (int c = 0; c < cn; ++c) {
        if (!((alive >> c) & 1u)) continue;
        float4 bc = chunkBox[c];
        float  ac = chunkArea[c];
        for (int d2 = c + 1; d2 < cn; ++d2) {
          if (!((alive >> d2) & 1u)) continue;
          float4 bd = chunkBox[d2];
          float iw = fminf(bc.z, bd.z) - fmaxf(bc.x, bd.x); iw = fmaxf(iw, 0.f);
          float ih = fminf(bc.w, bd.w) - fmaxf(bc.y, bd.y); ih = fmaxf(ih, 0.f);
          float inter = iw * ih;
          float denom = ac + chunkArea[d2] - inter + 1e-9f;
          if (inter > NMS_T * denom) alive &= ~(1u << d2);
        }
      }
      sh_alive = alive;
      int kc = sh_keepCount;
      for (int c = 0; c < cn && kc < POST_NMS; ++c)
        if ((alive >> c) & 1u)
          keepOut[(size_t)b * POST_NMS + (kc++)] = sidxs[chunkPos[c]];
      sh_keepCount = kc;
    }
    __syncthreads();
    if (__builtin_amdgcn_readfirstlane(sh_keepCount) >= POST_NMS) break;

    // suppress the remainder [start, N) against alive chunk members.
    // 16x16 tiles, one tile per wave per step. Pairwise area-sum via WMMA:
    //   D[m][n] = area_row[m]*1 + 1*area_col[n]   (rank-2 outer product)
    unsigned alive = (unsigned)__builtin_amdgcn_readfirstlane((int)sh_alive);
    int start      = __builtin_amdgcn_readfirstlane(sh_base);
    if (start < N && alive) {
      // A operand (16x4 f32): lanes 0-15 hold K0=area_row[m], K1=1; lanes 16-31 K2=K3=0
      v2f amat;
      amat.x = half ? 0.0f : chunkArea[ncol];
      amat.y = half ? 0.0f : 1.0f;
      // alive byte for this lane's 8 rows (rows half*8 .. half*8+7)
      unsigned rowMask = (alive >> (half << 3)) & 0xFFu;

      int firstTile = start >> 4;
      int lastTile  = (N - 1) >> 4;
      for (int t = firstTile + wv; t <= lastTile; t += 32) {
        int pos  = (t << 4) + ncol;
        int posC = pos < (N - 1) ? pos : (N - 1);   // clamped (branchless) LDS load
        float4 cb = boxesS[posC];
        float  ca = areaS[posC];

        // B operand (4x16 f32): lanes 0-15 hold K0=1, K1=area_col[n]; lanes 16-31 zero
        v2f bmat;
        bmat.x = half ? 0.0f : 1.0f;
        bmat.y = half ? 0.0f : ca;

        v8f cz = {};
        v8f dsum = __builtin_amdgcn_wmma_f32_16x16x4_f32(
            false, amat, false, bmat, (short)0, cz, false, false);

        unsigned supBits = 0u;
#pragma unroll
        for (int v = 0; v < 8; ++v) {
          int m = v + (half << 3);          // D: VGPR v -> row v (lanes 0-15) / v+8 (16-31)
          float4 rb = chunkBox[m];          // broadcast LDS load, unconditional
          float iw = fminf(rb.z, cb.z) - fmaxf(rb.x, cb.x); iw = fmaxf(iw, 0.f);
          float ih = fminf(rb.w, cb.w) - fmaxf(rb.y, cb.y); ih = fmaxf(ih, 0.f);
          float inter = iw * ih;
          float denom = dsum[v] - inter + 1e-9f;  // area_m + area_n - inter + eps
          supBits |= (inter > NMS_T * denom) ? (1u << v) : 0u;
        }
        if ((supBits & rowMask) && pos >= start && pos < N)
          atomicOr(&rm[pos >> 5], 1u << (pos & 31));
      }
    }
    __syncthreads();
  }

  // ---- pad keep list with -1 ----
  __syncthreads();
  for (int k2 = sh_keepCount + tid; k2 < POST_NMS; k2 += bs)
    keepOut[(size_t)b * POST_NMS + k2] = -1;
}

// ---------------------------------------------------------------------------
// Host launcher
// ---------------------------------------------------------------------------
extern "C" void kernel_launch(void* const* d_in, const int* in_sizes, int n_in,
                              void* d_out, int out_size, void* d_ws, size_t ws_size,
                              hipStream_t stream) {
  const float* scores = (const float*)d_in[0];
  const float* rois   = (const float*)d_in[1];
  const int*   ph     = (const int*)d_in[2];
  const int*   pw     = (const int*)d_in[3];

  int S = in_sizes[0];                       // B * N
  int B = (out_size - 4 * S) / POST_NMS;     // out = 4*B*N + B*POST_NMS
  if (B <= 0) B = 1;
  int N = S / B;

  float* outBoxes = (float*)d_out;
  int*   keepOut  = (int*)(outBoxes + (size_t)4 * S);

  int total = B * N;
  clip_kernel<<<(total + 255) / 256, 256, 0, stream>>>(rois, ph, pw, outBoxes, total);
  nms_kernel<<<B, NMS_THREADS, 0, stream>>>(scores, rois, ph, pw, keepOut, B, N);
}